// ResidualBlock_4277787427661
// MI455X (gfx1250) — compile-verified
//
#include <hip/hip_runtime.h>

typedef float v2f __attribute__((ext_vector_type(2)));
typedef float v8f __attribute__((ext_vector_type(8)));

#define N_NODES 100000
#define N_EDGES 1200000
#define DIM     64
#define BN_EPS  1e-5f

// ---------------------------------------------------------------------------
// 1) init: deg = 1.0 (self loop), BN accumulators = 0  (re-done every call)
// ---------------------------------------------------------------------------
__global__ void k_init(float* __restrict__ deg, float* __restrict__ bnacc) {
    int i = blockIdx.x * blockDim.x + threadIdx.x;
    if (i < N_NODES) deg[i] = 1.0f;
    if (i < 2 * DIM) bnacc[i] = 0.0f;
}

// ---------------------------------------------------------------------------
// 2) degree accumulation over dst (edge_index row 1), fp32 hw atomics
// ---------------------------------------------------------------------------
__global__ void k_deg(const long long* __restrict__ dst, float* __restrict__ deg) {
    int e = blockIdx.x * blockDim.x + threadIdx.x;
    if (e < N_EDGES) {
        unsafeAtomicAdd(&deg[(int)dst[e]], 1.0f);
    }
}

// ---------------------------------------------------------------------------
// 3) dinv = rsqrt(deg), in place
// ---------------------------------------------------------------------------
__global__ void k_rsqrt(float* __restrict__ deg) {
    int i = blockIdx.x * blockDim.x + threadIdx.x;
    if (i < N_NODES) deg[i] = rsqrtf(deg[i]);
}

// ---------------------------------------------------------------------------
// 4) xw = x @ W + b  via V_WMMA_F32_16X16X4_F32 (wave32, 16x16 tiles)
//    block = 256 threads = 8 waves; each block does a 128-row slab.
//    W is staged transposed in LDS with pad-stride 68 (conflict-free b64
//    reads: bank = (4n + k) % 64 distinct across the 16 lanes), x tile is
//    staged with the same padding. Bias is folded into the accumulator init
//    (C/D layout: each lane holds one output column n across its 8 VGPRs).
//    Block-uniform fullTile fast path: only the last of 782 blocks takes the
//    guarded tail, so no per-element exec-mask churn in the hot path.
// ---------------------------------------------------------------------------
#define GEMM_ROWS 128
#define LDS_PITCH 68

__launch_bounds__(256)
__global__ void k_gemm(const float* __restrict__ x, const float* __restrict__ W,
                       const float* __restrict__ bias, float* __restrict__ xw) {
    __shared__ float ldsW[DIM * LDS_PITCH];        // W transposed: ldsW[n*68 + k]
    __shared__ float ldsX[GEMM_ROWS * LDS_PITCH];  // ldsX[r*68 + k]

    const int tid  = threadIdx.x;
    const int lane = tid & 31;
    const int wave = tid >> 5;
    const int l15  = lane & 15;
    const int half = lane >> 4;                    // 0: K+0/1, 1: K+2/3 (ISA A layout)
    const long long rowBase = (long long)blockIdx.x * GEMM_ROWS;
    const bool fullTile = (rowBase + GEMM_ROWS) <= (long long)N_NODES;

    // stage W transposed (one-time strided LDS writes, coalesced global reads)
    for (int i = tid; i < DIM * DIM; i += 256) {
        int k = i >> 6, n = i & 63;
        ldsW[n * LDS_PITCH + k] = W[i];
    }
    // stage 128x64 x-tile, coalesced float4 global reads
    if (fullTile) {
        #pragma unroll
        for (int i = 0; i < 8; ++i) {
            int flat = i * 1024 + tid * 4;
            int r = flat >> 6, c = flat & 63;
            *(float4*)(&ldsX[r * LDS_PITCH + c]) =
                *(const float4*)(x + (rowBase + r) * DIM + c);
        }
    } else {
        #pragma unroll
        for (int i = 0; i < 8; ++i) {
            int flat = i * 1024 + tid * 4;
            int r = flat >> 6, c = flat & 63;
            long long grow = rowBase + r;
            float4 v = make_float4(0.f, 0.f, 0.f, 0.f);
            if (grow < N_NODES) v = *(const float4*)(x + grow * DIM + c);
            *(float4*)(&ldsX[r * LDS_PITCH + c]) = v;
        }
    }
    __syncthreads();

    const int waveRow = wave * 16;
    // Preload all 16 A fragments (each lane: half a row of x, as float2 pairs)
    v2f afrag[16];
    {
        const float* arow = &ldsX[(waveRow + l15) * LDS_PITCH];
        #pragma unroll
        for (int t = 0; t < 16; ++t) {
            int kk = t * 4 + half * 2;             // even -> 8B aligned
            afrag[t] = *(const v2f*)(arow + kk);
        }
    }
    // Per-lane bias values (one output column per lane per N-tile)
    float bias4[4];
    #pragma unroll
    for (int nt = 0; nt < 4; ++nt) bias4[nt] = bias[nt * 16 + l15];

    // 4 N-tiles of 16 columns; 16 WMMAs each over K=64, C seeded with bias
    #pragma unroll
    for (int nt = 0; nt < 4; ++nt) {
        const int n = nt * 16 + l15;
        const float bn = bias4[nt];
        v8f acc = {bn, bn, bn, bn, bn, bn, bn, bn};
        #pragma unroll
        for (int t = 0; t < 16; ++t) {
            int kk = t * 4 + half * 2;
            v2f bfrag = *(const v2f*)(&ldsW[n * LDS_PITCH + kk]);
            acc = __builtin_amdgcn_wmma_f32_16x16x4_f32(
                /*neg_a=*/false, afrag[t], /*neg_b=*/false, bfrag,
                /*c_mod=*/(short)0, acc, /*reuse_a=*/false, /*reuse_b=*/false);
        }
        // C/D layout: VGPR j -> row j (lanes 0-15) / row j+8 (lanes 16-31)
        if (fullTile) {
            #pragma unroll
            for (int j = 0; j < 8; ++j) {
                long long grow = rowBase + waveRow + j + half * 8;
                xw[grow * DIM + n] = acc[j];
            }
        } else {
            #pragma unroll
            for (int j = 0; j < 8; ++j) {
                long long grow = rowBase + waveRow + j + half * 8;
                if (grow < N_NODES) xw[grow * DIM + n] = acc[j];
            }
        }
    }
}

// ---------------------------------------------------------------------------
// 5) self-loop term: agg[i] = xw[i] * dinv[i]^2   (16 lanes / node, float4)
// ---------------------------------------------------------------------------
__global__ void k_self(const float* __restrict__ xw, const float* __restrict__ dinv,
                       float* __restrict__ agg) {
    long long t = (long long)blockIdx.x * blockDim.x + threadIdx.x;
    long long i = t >> 4;
    int c = (int)(t & 15) * 4;
    if (i < N_NODES) {
        float s = dinv[i]; s = s * s;
        float4 v = *(const float4*)(xw + i * DIM + c);
        v.x *= s; v.y *= s; v.z *= s; v.w *= s;
        *(float4*)(agg + i * DIM + c) = v;
    }
}

// ---------------------------------------------------------------------------
// 6) edge aggregation: agg[dst] += xw[src] * dinv[src]*dinv[dst]
//    16 lanes per edge, float4 gather, hw fp32 atomics (L2-resident RMW)
// ---------------------------------------------------------------------------
__global__ void k_edge(const long long* __restrict__ ei, const float* __restrict__ xw,
                       const float* __restrict__ dinv, float* __restrict__ agg) {
    long long t = (long long)blockIdx.x * blockDim.x + threadIdx.x;
    long long e = t >> 4;
    int c = (int)(t & 15) * 4;
    if (e < N_EDGES) {
        int s = (int)ei[e];
        int d = (int)ei[N_EDGES + e];
        float nrm = dinv[s] * dinv[d];
        float4 v = *(const float4*)(xw + (long long)s * DIM + c);
        float* ap = agg + (long long)d * DIM + c;
        unsafeAtomicAdd(ap + 0, v.x * nrm);
        unsafeAtomicAdd(ap + 1, v.y * nrm);
        unsafeAtomicAdd(ap + 2, v.z * nrm);
        unsafeAtomicAdd(ap + 3, v.w * nrm);
    }
}

// ---------------------------------------------------------------------------
// 7) BN stats: per-channel sum & sumsq over 100K rows
// ---------------------------------------------------------------------------
__launch_bounds__(256)
__global__ void k_bnstats(const float* __restrict__ agg, float* __restrict__ bnacc) {
    __shared__ float s0[256];
    __shared__ float s1[256];
    const int c  = threadIdx.x & 63;
    const int rl = threadIdx.x >> 6;               // 0..3
    const long long base = (long long)blockIdx.x * 256;
    float sum = 0.f, sq = 0.f;
    for (int r = rl; r < 256; r += 4) {
        long long row = base + r;
        if (row < N_NODES) {
            float v = agg[row * DIM + c];
            sum += v; sq += v * v;
        }
    }
    s0[threadIdx.x] = sum; s1[threadIdx.x] = sq;
    __syncthreads();
    if (rl == 0) {
        sum = s0[c] + s0[64 + c] + s0[128 + c] + s0[192 + c];
        sq  = s1[c] + s1[64 + c] + s1[128 + c] + s1[192 + c];
        unsafeAtomicAdd(&bnacc[c], sum);
        unsafeAtomicAdd(&bnacc[64 + c], sq);
    }
}

// ---------------------------------------------------------------------------
// 8) finalize: BN (biased var) + ReLU + residual + ReLU
// ---------------------------------------------------------------------------
__global__ void k_final(const float* __restrict__ agg, const float* __restrict__ x,
                        const float* __restrict__ gamma, const float* __restrict__ beta,
                        const float* __restrict__ bnacc, float* __restrict__ out) {
    long long t = (long long)blockIdx.x * blockDim.x + threadIdx.x;
    if (t < (long long)N_NODES * DIM) {
        int c = (int)(t & 63);
        const float invN = 1.0f / (float)N_NODES;
        float mean = bnacc[c] * invN;
        float var  = bnacc[64 + c] * invN - mean * mean;
        float inv  = rsqrtf(var + BN_EPS);
        float h = gamma[c] * (agg[t] - mean) * inv + beta[c];
        h = fmaxf(h, 0.0f);
        out[t] = fmaxf(h + x[t], 0.0f);
    }
}

// ---------------------------------------------------------------------------
extern "C" void kernel_launch(void* const* d_in, const int* in_sizes, int n_in,
                              void* d_out, int out_size, void* d_ws, size_t ws_size,
                              hipStream_t stream) {
    const float*     x     = (const float*)d_in[0];
    const long long* ei    = (const long long*)d_in[1];   // int64 [2, E]: row0=src, row1=dst
    const float*     W     = (const float*)d_in[2];
    const float*     bias  = (const float*)d_in[3];
    const float*     gamma = (const float*)d_in[4];
    const float*     beta  = (const float*)d_in[5];
    float*           out   = (float*)d_out;

    // workspace layout (floats): deg/dinv | xw | agg | bn accumulators
    float* ws    = (float*)d_ws;
    float* deg   = ws;                                   // N
    float* xw    = deg + N_NODES;                        // N*64
    float* agg   = xw + (size_t)N_NODES * DIM;           // N*64
    float* bnacc = agg + (size_t)N_NODES * DIM;          // 128

    k_init  <<<(N_NODES + 255) / 256, 256, 0, stream>>>(deg, bnacc);
    k_deg   <<<(N_EDGES + 255) / 256, 256, 0, stream>>>(ei + N_EDGES, deg);
    k_rsqrt <<<(N_NODES + 255) / 256, 256, 0, stream>>>(deg);
    k_gemm  <<<(N_NODES + GEMM_ROWS - 1) / GEMM_ROWS, 256, 0, stream>>>(x, W, bias, xw);
    k_self  <<<(int)(((long long)N_NODES * 16 + 255) / 256), 256, 0, stream>>>(xw, deg, agg);
    k_edge  <<<(int)(((long long)N_EDGES * 16 + 255) / 256), 256, 0, stream>>>(ei, xw, deg, agg);
    k_bnstats<<<(N_NODES + 255) / 256, 256, 0, stream>>>(agg, bnacc);
    k_final <<<(int)(((long long)N_NODES * DIM + 255) / 256), 256, 0, stream>>>(agg, x, gamma, beta, bnacc, out);
}